// SIGIR_21199958573600
// MI455X (gfx1250) — compile-verified
//
#include <hip/hip_runtime.h>

// ---------------------------------------------------------------------------
// SIGIR POI model for MI455X (gfx1250, wave32).
// All GEMMs run on V_WMMA_F32_16X16X4_F32 (exact f32 matrix cores).
// The big attn GEMM (1600x10000x128) fuses exp(-dist/range)*value_w[l] and the
// reduction over l; per-lane contributions with equal batch index are folded
// locally so each lane emits at most 2 global f32 atomics (not 8).
// ---------------------------------------------------------------------------

typedef __attribute__((ext_vector_type(2))) float v2f;
typedef __attribute__((ext_vector_type(8))) float v8f;

#define FEAT 128
#define LEN  25
#define BATCH 64
#define MROWS (BATCH * LEN)   // 1600
#define POI_N 10000
#define CAT_N 400

// ---------------- embeddings: x = sum of 4 gathers, xc likewise -------------
__global__ void k_embed(const int* __restrict__ user, const int* __restrict__ poi,
                        const int* __restrict__ cat, const int* __restrict__ tod,
                        const int* __restrict__ dow,
                        const float* __restrict__ user_emb, const float* __restrict__ poi_emb,
                        const float* __restrict__ tod_emb, const float* __restrict__ dow_emb,
                        const float* __restrict__ user_embc, const float* __restrict__ cat_emb,
                        const float* __restrict__ tod_embc, const float* __restrict__ dow_embc,
                        float* __restrict__ x, float* __restrict__ xc) {
  const int m = blockIdx.x;           // 0..1599
  const int d = threadIdx.x;          // 0..127
  const int u = user[m], p = poi[m], c = cat[m], t = tod[m], w = dow[m];
  x [(size_t)m * FEAT + d] = user_emb [(size_t)u * FEAT + d] + poi_emb[(size_t)p * FEAT + d]
                           + tod_emb  [(size_t)t * FEAT + d] + dow_emb[(size_t)w * FEAT + d];
  xc[(size_t)m * FEAT + d] = user_embc[(size_t)u * FEAT + d] + cat_emb[(size_t)c * FEAT + d]
                           + tod_embc [(size_t)t * FEAT + d] + dow_embc[(size_t)w * FEAT + d];
}

// ------------- global max/min of distance_matrix[poi] (two stage) -----------
__global__ void k_row_minmax(const int* __restrict__ poi, const float* __restrict__ dist,
                             float* __restrict__ pmin, float* __restrict__ pmax) {
  const int m = blockIdx.x;                           // one gathered row per block
  const float* row = dist + (size_t)poi[m] * POI_N;
  float lo = 3.4e38f, hi = -3.4e38f;
  for (int i = threadIdx.x; i < POI_N; i += blockDim.x) {
    float v = row[i];
    lo = fminf(lo, v); hi = fmaxf(hi, v);
  }
  __shared__ float smin[256], smax[256];
  smin[threadIdx.x] = lo; smax[threadIdx.x] = hi;
  __syncthreads();
  for (int st = 128; st > 0; st >>= 1) {
    if (threadIdx.x < st) {
      smin[threadIdx.x] = fminf(smin[threadIdx.x], smin[threadIdx.x + st]);
      smax[threadIdx.x] = fmaxf(smax[threadIdx.x], smax[threadIdx.x + st]);
    }
    __syncthreads();
  }
  if (threadIdx.x == 0) { pmin[m] = smin[0]; pmax[m] = smax[0]; }
}

__global__ void k_range(const float* __restrict__ pmin, const float* __restrict__ pmax,
                        float* __restrict__ invr, int n) {
  __shared__ float smin[256], smax[256];
  float lo = 3.4e38f, hi = -3.4e38f;
  for (int i = threadIdx.x; i < n; i += 256) {
    lo = fminf(lo, pmin[i]); hi = fmaxf(hi, pmax[i]);
  }
  smin[threadIdx.x] = lo; smax[threadIdx.x] = hi;
  __syncthreads();
  for (int st = 128; st > 0; st >>= 1) {
    if (threadIdx.x < st) {
      smin[threadIdx.x] = fminf(smin[threadIdx.x], smin[threadIdx.x + st]);
      smax[threadIdx.x] = fmaxf(smax[threadIdx.x], smax[threadIdx.x + st]);
    }
    __syncthreads();
  }
  if (threadIdx.x == 0) invr[0] = 1.0f / (smax[0] - smin[0]);
}

// ---------------- f32 WMMA GEMM:  C[m,n] = sum_k A[m,k]*Wt[n,k] + bias[n] ----
// One wave per 16x16 tile; K looped in steps of 4 -> v_wmma_f32_16x16x4_f32.
// A frag: lane&15 = M row, (lane>>4)*2 selects K pair (per ISA 16x4 layout).
// B frag: lane&15 = N col, (lane>>4)*2 selects K pair (transpose-symmetric).
// C/D:    VGPR r -> M = r + 8*(lane>>4), N = lane&15.
__global__ void k_gemm16(const float* __restrict__ A, const float* __restrict__ Wt,
                         const float* __restrict__ bias, float* __restrict__ C,
                         int K, int N) {
  const int lane = threadIdx.x;
  const int m0 = blockIdx.x * 16, n0 = blockIdx.y * 16;
  const int half = lane >> 4, id16 = lane & 15;
  const float* arow = A  + (size_t)(m0 + id16) * K + 2 * half;
  const float* brow = Wt + (size_t)(n0 + id16) * K + 2 * half;
  v8f acc = {};
#pragma unroll 8
  for (int kk = 0; kk < K; kk += 4) {
    v2f a = *(const v2f*)(arow + kk);
    v2f b = *(const v2f*)(brow + kk);
    acc = __builtin_amdgcn_wmma_f32_16x16x4_f32(false, a, false, b, (short)0, acc,
                                                false, false);
  }
  const float bval = bias ? bias[n0 + id16] : 0.0f;
#pragma unroll
  for (int r = 0; r < 8; ++r) {
    const int m = m0 + r + 8 * half;
    C[(size_t)m * N + n0 + id16] = acc[r] + bval;
  }
}

// ---- mean_value[b,tau] = (1/128) sum_d sum_s q[b,(s+tau)%L,d]*k[b,s,d] -----
__global__ void k_meanvalue(const float* __restrict__ q, const float* __restrict__ k,
                            float* __restrict__ meanv) {
  const int b = blockIdx.x / LEN, tau = blockIdx.x % LEN;
  const int d = threadIdx.x;
  const float* qb = q + (size_t)b * LEN * FEAT;
  const float* kb = k + (size_t)b * LEN * FEAT;
  float s = 0.0f;
#pragma unroll
  for (int t = 0; t < LEN; ++t) {
    int tq = t + tau; if (tq >= LEN) tq -= LEN;
    s += qb[tq * FEAT + d] * kb[t * FEAT + d];
  }
  __shared__ float red[FEAT];
  red[d] = s;
  __syncthreads();
  for (int st = 64; st > 0; st >>= 1) {
    if (d < st) red[d] += red[d + st];
    __syncthreads();
  }
  if (d == 0) meanv[blockIdx.x] = red[0] * (1.0f / 128.0f);
}

// ----- top-3 of batch-mean, then per-batch 3-way softmax of mean_value ------
__global__ void k_topk_softmax(const float* __restrict__ meanv,
                               int* __restrict__ idx, float* __restrict__ wsm) {
  __shared__ float g[LEN];
  __shared__ int sidx[3];
  const int t = threadIdx.x;
  if (t < LEN) {
    float s = 0.0f;
    for (int b = 0; b < BATCH; ++b) s += meanv[b * LEN + t];
    g[t] = s * (1.0f / BATCH);
  }
  __syncthreads();
  if (t == 0) {
    bool used[LEN];
    for (int i = 0; i < LEN; ++i) used[i] = false;
    for (int j = 0; j < 3; ++j) {
      int best = 0; float bv = -3.4e38f;
      for (int i = 0; i < LEN; ++i)
        if (!used[i] && g[i] > bv) { bv = g[i]; best = i; }
      used[best] = true; sidx[j] = best; idx[j] = best;
    }
  }
  __syncthreads();
  if (t < BATCH) {
    const float w0 = meanv[t * LEN + sidx[0]];
    const float w1 = meanv[t * LEN + sidx[1]];
    const float w2 = meanv[t * LEN + sidx[2]];
    const float mx = fmaxf(w0, fmaxf(w1, w2));
    const float e0 = __expf(w0 - mx), e1 = __expf(w1 - mx), e2 = __expf(w2 - mx);
    const float inv = 1.0f / (e0 + e1 + e2);
    wsm[t * 3 + 0] = e0 * inv; wsm[t * 3 + 1] = e1 * inv; wsm[t * 3 + 2] = e2 * inv;
  }
}

// --------- y[b,l,d] = sum_j wsm[b,j] * v[b,(l+idx[j])%L,d] ------------------
__global__ void k_combine(const float* __restrict__ v, const int* __restrict__ idx,
                          const float* __restrict__ wsm, float* __restrict__ y) {
  const int m = blockIdx.x, d = threadIdx.x;
  const int b = m / LEN, l = m % LEN;
  const float* vb = v + (size_t)b * LEN * FEAT;
  float acc = 0.0f;
#pragma unroll
  for (int j = 0; j < 3; ++j) {
    int ls = l + idx[j]; if (ls >= LEN) ls -= LEN;
    acc += wsm[b * 3 + j] * vb[ls * FEAT + d];
  }
  y[(size_t)m * FEAT + d] = acc;
}

// ------------------------- output init to value_b ---------------------------
__global__ void k_init_out(float* __restrict__ out, int n, const float* __restrict__ vb) {
  const int i = blockIdx.x * blockDim.x + threadIdx.x;
  if (i < n) out[i] = vb[0];
}

// -------- fused attention: S = X @ E^T (WMMA), epilogue folds
//          value_w[l] * (dist ? exp(-dist[poi[m],n]*invr) : 1); contributions
//          over l with the same batch b are summed in-register, so each lane
//          issues at most 2 global f32 atomics for its 8 output rows. --------
__global__ void k_attn_wmma(const float* __restrict__ X, const float* __restrict__ E,
                            int N, const float* __restrict__ vw,
                            const int* __restrict__ poiIdx, const float* __restrict__ dist,
                            const float* __restrict__ invr, float* __restrict__ outp) {
  const int lane = threadIdx.x;
  const int m0 = blockIdx.x * 16, n0 = blockIdx.y * 16;
  const int half = lane >> 4, id16 = lane & 15;
  const float* arow = X + (size_t)(m0 + id16) * FEAT + 2 * half;
  const float* brow = E + (size_t)(n0 + id16) * FEAT + 2 * half;
  v8f acc = {};
#pragma unroll 8
  for (int kk = 0; kk < FEAT; kk += 4) {
    v2f a = *(const v2f*)(arow + kk);
    v2f b = *(const v2f*)(brow + kk);
    acc = __builtin_amdgcn_wmma_f32_16x16x4_f32(false, a, false, b, (short)0, acc,
                                                false, false);
  }
  const int n = n0 + id16;
  const float ir = invr ? invr[0] : 0.0f;
  float pend = 0.0f;
  int pendb = -1;
#pragma unroll
  for (int r = 0; r < 8; ++r) {
    const int m = m0 + r + 8 * half;   // 8 consecutive m rows per lane
    const int b = m / LEN, l = m - b * LEN;
    float contrib = vw[l] * acc[r];
    if (dist)
      contrib *= __expf(-dist[(size_t)poiIdx[m] * POI_N + n] * ir);
    if (b != pendb) {                  // crosses at most one batch boundary
      if (pendb >= 0) atomicAdd(&outp[(size_t)pendb * N + n], pend);
      pendb = b; pend = contrib;
    } else {
      pend += contrib;
    }
  }
  atomicAdd(&outp[(size_t)pendb * N + n], pend);
}

// ---------------------------------------------------------------------------
static void ac_layer(hipStream_t stream, const float* qin, const float* kvin,
                     const float* Wsub, const float* Bsub, float* dst,
                     float* qb, float* kb, float* vb, float* yb,
                     float* meanv, int* idxb, float* wsm) {
  const int F = FEAT;
  dim3 g(MROWS / 16, F / 16);
  k_gemm16<<<g, 32, 0, stream>>>(qin,  Wsub + 0 * F * F, Bsub + 0 * F, qb, F, F);
  k_gemm16<<<g, 32, 0, stream>>>(kvin, Wsub + 1 * F * F, Bsub + 1 * F, kb, F, F);
  k_gemm16<<<g, 32, 0, stream>>>(kvin, Wsub + 2 * F * F, Bsub + 2 * F, vb, F, F);
  k_meanvalue<<<MROWS, F, 0, stream>>>(qb, kb, meanv);
  k_topk_softmax<<<1, 64, 0, stream>>>(meanv, idxb, wsm);
  k_combine<<<MROWS, F, 0, stream>>>(vb, idxb, wsm, yb);
  k_gemm16<<<g, 32, 0, stream>>>(yb, Wsub + 3 * F * F, Bsub + 3 * F, dst, F, F);
}

extern "C" void kernel_launch(void* const* d_in, const int* in_sizes, int n_in,
                              void* d_out, int out_size, void* d_ws, size_t ws_size,
                              hipStream_t stream) {
  (void)in_sizes; (void)n_in; (void)ws_size;
  const int F = FEAT;

  const int*   user      = (const int*)  d_in[0];
  const int*   poi       = (const int*)  d_in[1];
  const int*   cat       = (const int*)  d_in[2];
  const int*   tod       = (const int*)  d_in[5];
  const int*   dow       = (const int*)  d_in[6];
  const float* user_emb  = (const float*)d_in[8];
  const float* poi_emb   = (const float*)d_in[9];
  const float* cat_emb   = (const float*)d_in[10];
  const float* tod_emb   = (const float*)d_in[11];
  const float* dow_emb   = (const float*)d_in[12];
  const float* user_embc = (const float*)d_in[13];
  const float* tod_embc  = (const float*)d_in[14];
  const float* dow_embc  = (const float*)d_in[15];
  const float* Wl        = (const float*)d_in[16];
  const float* Bvl       = (const float*)d_in[17];
  const float* value_w   = (const float*)d_in[18];
  const float* value_b   = (const float*)d_in[19];
  const float* dist      = (const float*)d_in[20];

  // ---- workspace carving (floats) ----
  float* ws = (float*)d_ws;
  size_t o = 0;
  const size_t ACT = (size_t)MROWS * F;
  float* act0 = ws + o; o += ACT;       // out chain
  float* act1 = ws + o; o += ACT;       // outc chain
  float* act2 = ws + o; o += ACT;       // temp
  float* qb   = ws + o; o += ACT;
  float* kb   = ws + o; o += ACT;
  float* vb   = ws + o; o += ACT;
  float* yb   = ws + o; o += ACT;
  float* meanv = ws + o; o += MROWS;
  float* wsm   = ws + o; o += (size_t)BATCH * 3 + 1;
  float* pmin  = ws + o; o += MROWS;
  float* pmax  = ws + o; o += MROWS;
  float* invr  = ws + o; o += 4;
  int*   idxb  = (int*)(ws + o); o += 8;

  // ---- embeddings ----
  k_embed<<<MROWS, F, 0, stream>>>(user, poi, cat, tod, dow,
                                   user_emb, poi_emb, tod_emb, dow_emb,
                                   user_embc, cat_emb, tod_embc, dow_embc,
                                   act0, act1);

  // ---- interval range (max-min over gathered rows) ----
  k_row_minmax<<<MROWS, 256, 0, stream>>>(poi, dist, pmin, pmax);
  k_range<<<1, 256, 0, stream>>>(pmin, pmax, invr, MROWS);

  // ---- 2 layers x 4 cross-wired AC sub-layers ----
  float* pOut  = act0;
  float* pOutc = act1;
  float* pTmp  = act2;
  for (int i = 0; i < 2; ++i) {
    const float* Wi = Wl  + (size_t)i * 4 * 4 * F * F;
    const float* Bi = Bvl + (size_t)i * 4 * 4 * F;
    // out = AC(out, out)
    ac_layer(stream, pOut, pOut, Wi + 0 * (size_t)4 * F * F, Bi + 0 * 4 * F,
             pTmp, qb, kb, vb, yb, meanv, idxb, wsm);
    { float* t = pOut; pOut = pTmp; pTmp = t; }
    // outc = AC(outc, outc)
    ac_layer(stream, pOutc, pOutc, Wi + 1 * (size_t)4 * F * F, Bi + 1 * 4 * F,
             pTmp, qb, kb, vb, yb, meanv, idxb, wsm);
    { float* t = pOutc; pOutc = pTmp; pTmp = t; }
    // out = AC(out, outc)
    ac_layer(stream, pOut, pOutc, Wi + 2 * (size_t)4 * F * F, Bi + 2 * 4 * F,
             pTmp, qb, kb, vb, yb, meanv, idxb, wsm);
    { float* t = pOut; pOut = pTmp; pTmp = t; }
    // outc = AC(outc, out)
    ac_layer(stream, pOutc, pOut, Wi + 3 * (size_t)4 * F * F, Bi + 3 * 4 * F,
             pTmp, qb, kb, vb, yb, meanv, idxb, wsm);
    { float* t = pOutc; pOutc = pTmp; pTmp = t; }
  }

  // ---- outputs: pre_poi (64x10000) then pre_cat (64x400), init to value_b --
  float* outp = (float*)d_out;
  k_init_out<<<(out_size + 255) / 256, 256, 0, stream>>>(outp, out_size, value_b);

  dim3 gp(MROWS / 16, POI_N / 16);
  k_attn_wmma<<<gp, 32, 0, stream>>>(pOut, poi_emb, POI_N, value_w,
                                     poi, dist, invr, outp);
  dim3 gc(MROWS / 16, CAT_N / 16);
  k_attn_wmma<<<gc, 32, 0, stream>>>(pOutc, cat_emb, CAT_N, value_w,
                                     nullptr, nullptr, nullptr,
                                     outp + (size_t)BATCH * POI_N);
}